// ChamferDistance_24464133718446
// MI455X (gfx1250) — compile-verified
//
#include <hip/hip_runtime.h>

// CDNA5 wave32 WMMA operand types
typedef __attribute__((ext_vector_type(2))) float v2f;  // 16x4 f32 A / 4x16 f32 B: 2 VGPRs/lane
typedef __attribute__((ext_vector_type(8))) float v8f;  // 16x16 f32 C/D: 8 VGPRs/lane

// Raw v_min3_num_f32: 3-way float min in one VALU op, no NaN-canonicalize
// (inputs here are always finite).
__device__ __forceinline__ float min3f(float a, float b, float c) {
    float r;
    asm("v_min3_num_f32 %0, %1, %2, %3" : "=v"(r) : "v"(a), "v"(b), "v"(c));
    return r;
}

// Pack (x,y,z) -> (x,y,z, x^2+y^2+z^2): one 128-bit load per point afterwards,
// and the (x,y)/(z,|p|^2) halves line up with the WMMA B-operand lane split.
__global__ void pack_pts_kernel(const float* __restrict__ xyz,
                                float4* __restrict__ out, int n) {
    int i = blockIdx.x * blockDim.x + threadIdx.x;
    if (i < n) {
        float x = xyz[3 * i + 0];
        float y = xyz[3 * i + 1];
        float z = xyz[3 * i + 2];
        out[i] = make_float4(x, y, z, x * x + y * y + z * z);
    }
}

// One wave = 16 query rows. A[m][:] = (-2qx, -2qy, -2qz, 1); B[:][n] = (bx,by,bz,|b|^2).
// D = A*B = -2 q.b + |b|^2 per pair; running min over db columns via v_min3_num,
// add |q|^2 in the epilogue. K=4 maps exactly onto V_WMMA_F32_16X16X4_F32.
__global__ void __launch_bounds__(256) chamfer_min_wmma_kernel(
    const float4* __restrict__ q4, const float4* __restrict__ db4,
    float* __restrict__ out, int nQ, int nDb) {
    const int lane   = threadIdx.x & 31;
    const int waveId = threadIdx.x >> 5;
    const int batch  = blockIdx.y;

    q4  += (size_t)batch * nQ;
    db4 += (size_t)batch * nDb;
    out += (size_t)batch * nQ;

    const int  qbase = (blockIdx.x * 8 + waveId) * 16;  // 16 rows per wave
    const int  lrow  = lane & 15;
    const bool hi    = lane >= 16;  // lanes 16-31 carry K=2,3 (A) / rows K=2,3 (B)

    // A-matrix: 16x4 f32, ISA layout. Lane<16: (K=0,K=1) = (-2qx,-2qy);
    // lane>=16: (K=2,K=3) = (-2qz, 1).
    const float4 q = q4[qbase + lrow];
    v2f a;
    a.x = hi ? (-2.0f * q.z) : (-2.0f * q.x);
    a.y = hi ? 1.0f          : (-2.0f * q.y);

    const float INF = __builtin_inff();
    v8f acc = {INF, INF, INF, INF, INF, INF, INF, INF};
    const v8f zero = {0.f, 0.f, 0.f, 0.f, 0.f, 0.f, 0.f, 0.f};

    // Each lane loads exactly its B half directly: lane<16 -> (x,y) [K=0,1],
    // lane>=16 -> (z,|b|^2) [K=2,3]. Single global_load_b64, no selects.
    const float2* db2 = (const float2*)db4;
    const int lidx = lrow * 2 + (hi ? 1 : 0);

    // 64 db points per iteration: 4 independent B tiles -> 4 independent WMMAs.
#pragma unroll 2
    for (int t = 0; t < nDb; t += 64) {
        const int base = t * 2 + lidx;
        const float2 p0 = db2[base + 0];
        const float2 p1 = db2[base + 32];
        const float2 p2 = db2[base + 64];
        const float2 p3 = db2[base + 96];

        v2f b0; b0.x = p0.x; b0.y = p0.y;
        v2f b1; b1.x = p1.x; b1.y = p1.y;
        v2f b2; b2.x = p2.x; b2.y = p2.y;
        v2f b3; b3.x = p3.x; b3.y = p3.y;

        // (neg_a, A, neg_b, B, c_mod, C, reuse_a, reuse_b)
        v8f d0 = __builtin_amdgcn_wmma_f32_16x16x4_f32(false, a, false, b0, (short)0, zero, false, false);
        v8f d1 = __builtin_amdgcn_wmma_f32_16x16x4_f32(false, a, false, b1, (short)0, zero, false, false);
        v8f d2 = __builtin_amdgcn_wmma_f32_16x16x4_f32(false, a, false, b2, (short)0, zero, false, false);
        v8f d3 = __builtin_amdgcn_wmma_f32_16x16x4_f32(false, a, false, b3, (short)0, zero, false, false);

        // 2 x v_min3_num_f32 per component: fold 4 tiles + accumulator.
#pragma unroll
        for (int c = 0; c < 8; ++c) {
            float m  = min3f(d0[c], d1[c], d2[c]);
            acc[c]   = min3f(acc[c], m, d3[c]);
        }
    }

    // Min-reduce each C/D VGPR across its 16-lane half
    // (C layout: VGPR c = row c in lanes 0-15, row c+8 in lanes 16-31).
    float red[8];
#pragma unroll
    for (int c = 0; c < 8; ++c) {
        float r = acc[c];
        r = fminf(r, __shfl_xor(r, 1, 32));
        r = fminf(r, __shfl_xor(r, 2, 32));
        r = fminf(r, __shfl_xor(r, 4, 32));
        r = fminf(r, __shfl_xor(r, 8, 32));
        red[c] = r;
    }

    if (lrow == 0) {
        const int rowoff = hi ? 8 : 0;
#pragma unroll
        for (int c = 0; c < 8; ++c) {
            const int row = qbase + rowoff + c;
            out[row] = red[c] + q4[row].w;  // add |q|^2
        }
    }
}

extern "C" void kernel_launch(void* const* d_in, const int* in_sizes, int n_in,
                              void* d_out, int out_size, void* d_ws, size_t ws_size,
                              hipStream_t stream) {
    const int B = 2, N = 32768, M = 32768;
    const float* xyz1 = (const float*)d_in[0];
    const float* xyz2 = (const float*)d_in[1];
    float* out = (float*)d_out;

    float4* ws1 = (float4*)d_ws;            // B*N packed points (1 MB)
    float4* ws2 = ws1 + (size_t)B * N;      // B*M packed points (1 MB)

    const int n1 = B * N, n2 = B * M;
    pack_pts_kernel<<<dim3((n1 + 255) / 256), dim3(256), 0, stream>>>(xyz1, ws1, n1);
    pack_pts_kernel<<<dim3((n2 + 255) / 256), dim3(256), 0, stream>>>(xyz2, ws2, n2);

    dim3 blk(256);                          // 8 waves -> 128 query rows / block
    dim3 g1(N / 128, B), g2(M / 128, B);
    // dir1: each xyz1 point vs all xyz2
    chamfer_min_wmma_kernel<<<g1, blk, 0, stream>>>(ws1, ws2, out, N, M);
    // dir2: each xyz2 point vs all xyz1
    chamfer_min_wmma_kernel<<<g2, blk, 0, stream>>>(ws2, ws1, out + (size_t)B * N, M, N);
}